// EmbeddingGrad_76364518523444
// MI455X (gfx1250) — compile-verified
//
#include <hip/hip_runtime.h>

// out[row, d] = W[d * NUM_EMB + idx[row]] + b[d]
//   idx : [204800] int32   (4096 x 50)
//   W   : [128, 100000] fp32 (51.2 MB -> resident in MI455X 192MB L2)
//   b   : [128] fp32
//   out : [204800, 128] fp32 (105 MB, write-once -> non-temporal stores)
//
// Memory-bound gather: ~157 MB of traffic @ 23.3 TB/s ~ 6.7 us.
// WMMA (one-hot matmul form) would be ~5.2 TFLOP -> ~1000x slower; omitted
// deliberately.

typedef float v4f __attribute__((ext_vector_type(4)));

#define NUM_EMB 100000
#define EMB_DIM 128

__global__ __launch_bounds__(256) void embed_gather_kernel(
    const int* __restrict__ idx,
    const float* __restrict__ W,
    const float* __restrict__ b,
    float* __restrict__ out,
    int n_rows)
{
    const int lane = threadIdx.x & 31;        // 32 lanes cooperate on one row
    const int d0   = lane << 2;               // 4 dims per lane -> b128 store
    const int rows_per_block = blockDim.x >> 5;               // 8
    const int row0 = blockIdx.x * rows_per_block + (threadIdx.x >> 5);
    const int row_stride = gridDim.x * rows_per_block;

    // Bias for this lane's fixed dim group: load once, keep in registers.
    const v4f b4 = *(const v4f*)(b + d0);

    // Per-lane base into W for dim d0 (32-bit offsets: W spans only 51.2 MB).
    const float* __restrict__ Wd = W + d0 * NUM_EMB;

    if (row0 >= n_rows) return;

    // Software-pipeline the index so we can prefetch next iteration's
    // scattered W lines (gfx1250 global_prefetch_b8) while current loads fly.
    int col = idx[row0];

    for (int row = row0; row < n_rows; row += row_stride) {
        const int nrow = row + row_stride;
        int ncol = col;
        if (nrow < n_rows) {
            ncol = idx[nrow];
            const float* pw = Wd + ncol;
            __builtin_prefetch(pw + 0 * NUM_EMB, 0, 3);
            __builtin_prefetch(pw + 1 * NUM_EMB, 0, 3);
            __builtin_prefetch(pw + 2 * NUM_EMB, 0, 3);
            __builtin_prefetch(pw + 3 * NUM_EMB, 0, 3);
        }

        // 4 scattered 4B loads (400KB stride between dims) -> serviced by L2,
        // since the whole W matrix is L2-resident after first touch.
        const float* wp = Wd + col;
        v4f v;
        v.x = wp[0 * NUM_EMB];
        v.y = wp[1 * NUM_EMB];
        v.z = wp[2 * NUM_EMB];
        v.w = wp[3 * NUM_EMB];

        v.x += b4.x;
        v.y += b4.y;
        v.z += b4.z;
        v.w += b4.w;

        // Coalesced non-temporal b128 store: whole wave writes one contiguous
        // 512B embedding row; NT keeps the 105MB output from evicting W in L2.
        __builtin_nontemporal_store(v, (v4f*)(out + row * EMB_DIM + d0));

        col = ncol;
    }
}

extern "C" void kernel_launch(void* const* d_in, const int* in_sizes, int n_in,
                              void* d_out, int out_size, void* d_ws, size_t ws_size,
                              hipStream_t stream) {
    const int*   idx = (const int*)  d_in[0];   // [4096*50] (int32 per harness)
    const float* W   = (const float*)d_in[1];   // [128, 100000]
    const float* b   = (const float*)d_in[2];   // [128]
    float*       out = (float*)      d_out;     // [4096*50, 128]

    const int n_rows = in_sizes[0];             // 204800
    const int rows_per_block = 256 / 32;        // 8
    int grid = (n_rows + rows_per_block - 1) / rows_per_block;
    if (grid > 4096) grid = 4096;               // grid-stride; saturates WGPs

    embed_gather_kernel<<<grid, 256, 0, stream>>>(idx, W, b, out, n_rows);
}